// knn_CrossLayerLight_25220047962584
// MI455X (gfx1250) — compile-verified
//
#include <hip/hip_runtime.h>

// ---------------------------------------------------------------------------
// CrossLayerLight for MI455X (gfx1250). All GEMMs use V_WMMA_F32_16X16X4_F32
// (exact f32 on the matrix pipe; problem is bandwidth/latency bound so f32 is
// the right precision and keeps top_k rankings faithful).
// ---------------------------------------------------------------------------

#define LRELU_SLOPE 0.1f

typedef float v2f __attribute__((ext_vector_type(2)));
typedef float v8f __attribute__((ext_vector_type(8)));

__device__ __forceinline__ float leaky(float x) { return x > 0.f ? x : LRELU_SLOPE * x; }

// Generic 16x16 f32 WMMA tile accumulate: C += A(MxK) * B(KxN).
// A(m,k) = A[m*Ar + k*Ac], B(k,n) = B[k*Br + n*Bc]. kdim multiple of 4.
// ISA layout (CDNA5 ISA): A 16x4 in 2 VGPRs (lanes<16: K0/K1, lanes>=16: K2/K3),
// B 4x16 in 2 VGPRs (VGPR0: K0 lo / K2 hi, VGPR1: K1 lo / K3 hi),
// C/D 8 VGPRs (VGPR v: M=v lanes<16, M=v+8 lanes>=16; N=lane&15).
__device__ __forceinline__ v8f wmma_tile_f32(const float* __restrict__ A, int Ar, int Ac,
                                             const float* __restrict__ B, int Br, int Bc,
                                             int m0, int n0, int kdim, v8f acc) {
  const int lane = threadIdx.x & 31;
  const int half = lane >> 4;
  const int l16  = lane & 15;
  const int arow = m0 + l16;
  const int bcol = n0 + l16;
  for (int k0 = 0; k0 < kdim; k0 += 4) {
    const int ka = k0 + 2 * half;
    v2f a, b;
    a.x = A[arow * Ar + ka * Ac];
    a.y = A[arow * Ar + (ka + 1) * Ac];
    b.x = B[ka * Br + bcol * Bc];
    b.y = B[(ka + 1) * Br + bcol * Bc];
    acc = __builtin_amdgcn_wmma_f32_16x16x4_f32(false, a, false, b, (short)0, acc,
                                                false, false);
  }
  return acc;
}

// out[o,n] = sum_c w[o,c] * x[c,n] + b[o]   (64 x N, K=64). 1 wave per 16x16 tile.
__global__ void k_conv64(const float* __restrict__ x, const float* __restrict__ w,
                         const float* __restrict__ b, float* __restrict__ out, int N) {
  const int wid = blockIdx.x * (blockDim.x >> 5) + (threadIdx.x >> 5);
  const int ntiles = N >> 4;
  const int mt = wid / ntiles, nt = wid - mt * ntiles;
  const int m0 = mt << 4, n0 = nt << 4;
  v8f acc = {};
  acc = wmma_tile_f32(w, 64, 1, x, N, 1, m0, n0, 64, acc);
  const int lane = threadIdx.x & 31, half = lane >> 4, l16 = lane & 15;
  const int col = n0 + l16;
#pragma unroll
  for (int v = 0; v < 8; ++v) {
    const int row = m0 + v + 8 * half;
    out[row * N + col] = acc[v] + b[row];
  }
}

// corr[n1,n2] = scale * sum_d f1[d,n1] * f2[d,n2]   (N x N, K=64)
__global__ void k_corr(const float* __restrict__ f1, const float* __restrict__ f2,
                       float* __restrict__ out, int N, float scale) {
  const int wid = blockIdx.x * (blockDim.x >> 5) + (threadIdx.x >> 5);
  const int ntiles = N >> 4;
  const int mt = wid / ntiles, nt = wid - mt * ntiles;
  const int m0 = mt << 4, n0 = nt << 4;
  v8f acc = {};
  acc = wmma_tile_f32(f1, 1, N, f2, N, 1, m0, n0, 64, acc);  // A = f1^T
  const int lane = threadIdx.x & 31, half = lane >> 4, l16 = lane & 15;
  const int col = n0 + l16;
#pragma unroll
  for (int v = 0; v < 8; ++v)
    out[(m0 + v + 8 * half) * N + col] = acc[v] * scale;
}

// Y[m,o] = leaky( sum_d X[m,d]*w[o,d] + b[o] )   (M x 64, K=64), B = w^T
__global__ void k_mlp64(const float* __restrict__ X, const float* __restrict__ w,
                        const float* __restrict__ b, float* __restrict__ Y, int M) {
  const int wid = blockIdx.x * (blockDim.x >> 5) + (threadIdx.x >> 5);
  const int mt = wid >> 2, nt = wid & 3;  // 64/16 = 4 column tiles
  const int m0 = mt << 4, n0 = nt << 4;
  v8f acc = {};
  acc = wmma_tile_f32(X, 64, 1, w, 1, 64, m0, n0, 64, acc);
  const int lane = threadIdx.x & 31, half = lane >> 4, l16 = lane & 15;
  const int col = n0 + l16;
#pragma unroll
  for (int v = 0; v < 8; ++v)
    Y[(m0 + v + 8 * half) * 64 + col] = leaky(acc[v] + b[col]);
  (void)M;
}

// total-order comparator: "a precedes b" in descending top_k order
__device__ __forceinline__ bool pair_gt(float va, int ia, float vb, int ib) {
  return (va > vb) || (va == vb && ia < ib);
}

// Per-row top-512 of 2048 via full bitonic sort in LDS (values + indices).
__global__ void k_top512(const float* __restrict__ corr, float* __restrict__ cv,
                         int* __restrict__ ci, int N, int K) {
  __shared__ float sv[2048];
  __shared__ int   si[2048];
  const int n = blockIdx.x;
  const float* row = corr + (long)n * N;
  for (int i = threadIdx.x; i < N; i += blockDim.x) { sv[i] = row[i]; si[i] = i; }
  __syncthreads();
  for (int k = 2; k <= N; k <<= 1) {
    for (int j = k >> 1; j > 0; j >>= 1) {
      for (int i = (int)threadIdx.x; i < N; i += blockDim.x) {
        const int ixj = i ^ j;
        if (ixj > i) {
          const float va = sv[i], vb = sv[ixj];
          const int ia = si[i], ib = si[ixj];
          const bool bFirst = pair_gt(vb, ib, va, ia);
          const bool doSwap = ((i & k) == 0) ? bFirst : !bFirst;  // final order descending
          if (doSwap) { sv[i] = vb; si[i] = ib; sv[ixj] = va; si[ixj] = ia; }
        }
      }
      __syncthreads();
    }
  }
  for (int i = threadIdx.x; i < K; i += blockDim.x) {
    cv[(long)n * K + i] = sv[i];
    ci[(long)n * K + i] = si[i];
  }
}

// WeightNet 4->32->16->8->1 (linear, linear, relu, relu-linear, relu-linear)
__global__ void k_weightnet(const float* __restrict__ xyzq, const float* __restrict__ xyzs,
                            const float* __restrict__ cv, const int* __restrict__ ci,
                            const float* __restrict__ w1, const float* __restrict__ b1,
                            const float* __restrict__ w2, const float* __restrict__ b2,
                            const float* __restrict__ w3, const float* __restrict__ b3,
                            const float* __restrict__ w4, const float* __restrict__ b4,
                            float* __restrict__ h, int N, int K) {
  __shared__ float s_w1[128], s_b1[32], s_w2[512], s_b2[16], s_w3[128], s_b3[8], s_w4[8], s_b4[1];
  for (int i = threadIdx.x; i < 128; i += blockDim.x) s_w1[i] = w1[i];
  for (int i = threadIdx.x; i < 32;  i += blockDim.x) s_b1[i] = b1[i];
  for (int i = threadIdx.x; i < 512; i += blockDim.x) s_w2[i] = w2[i];
  for (int i = threadIdx.x; i < 16;  i += blockDim.x) s_b2[i] = b2[i];
  for (int i = threadIdx.x; i < 128; i += blockDim.x) s_w3[i] = w3[i];
  for (int i = threadIdx.x; i < 8;   i += blockDim.x) s_b3[i] = b3[i];
  for (int i = threadIdx.x; i < 8;   i += blockDim.x) s_w4[i] = w4[i];
  if (threadIdx.x == 0) s_b4[0] = b4[0];
  __syncthreads();
  const int g = blockIdx.x * blockDim.x + threadIdx.x;
  if (g >= N * K) return;
  const int n = g / K;
  const int j = ci[g];
  const float f0 = xyzs[0 * N + j] - xyzq[0 * N + n];
  const float f1 = xyzs[1 * N + j] - xyzq[1 * N + n];
  const float f2 = xyzs[2 * N + j] - xyzq[2 * N + n];
  const float f3 = cv[g];
  float h1[32];
#pragma unroll
  for (int o = 0; o < 32; ++o)
    h1[o] = s_b1[o] + s_w1[o * 4 + 0] * f0 + s_w1[o * 4 + 1] * f1 +
            s_w1[o * 4 + 2] * f2 + s_w1[o * 4 + 3] * f3;
  float h2[16];
#pragma unroll
  for (int o = 0; o < 16; ++o) {
    float a = s_b2[o];
#pragma unroll
    for (int c = 0; c < 32; ++c) a += s_w2[o * 32 + c] * h1[c];
    h2[o] = fmaxf(a, 0.f);
  }
  float h3[8];
#pragma unroll
  for (int o = 0; o < 8; ++o) {
    float a = s_b3[o];
#pragma unroll
    for (int c = 0; c < 16; ++c) a += s_w3[o * 16 + c] * h2[c];
    h3[o] = fmaxf(a, 0.f);
  }
  float a = s_b4[0];
#pragma unroll
  for (int c = 0; c < 8; ++c) a += s_w4[c] * h3[c];
  h[g] = fmaxf(a, 0.f);
}

// softmax-over-N stats per column k: colmax[k], colsum[k] = sum_n exp(h[n,k]-max)
__global__ void k_colstats(const float* __restrict__ h, float* __restrict__ colmax,
                           float* __restrict__ colsum, int N, int K) {
  __shared__ float red[256];
  const int k = blockIdx.x;
  const int tid = threadIdx.x;
  float m = -3.402823466e+38f;
  for (int n = tid; n < N; n += blockDim.x) m = fmaxf(m, h[n * K + k]);
  red[tid] = m; __syncthreads();
  for (int s = blockDim.x >> 1; s > 0; s >>= 1) {
    if (tid < s) red[tid] = fmaxf(red[tid], red[tid + s]);
    __syncthreads();
  }
  const float cm = red[0]; __syncthreads();
  float sm = 0.f;
  for (int n = tid; n < N; n += blockDim.x) sm += expf(h[n * K + k] - cm);
  red[tid] = sm; __syncthreads();
  for (int s = blockDim.x >> 1; s > 0; s >>= 1) {
    if (tid < s) red[tid] += red[tid + s];
    __syncthreads();
  }
  if (tid == 0) { colmax[k] = cm; colsum[k] = red[0]; }
}

// Per-row top-32 of p[n,k] = exp(h-colmax[k])/colsum[k] over k in [0,512)
__global__ void k_top32(const float* __restrict__ h, const float* __restrict__ colmax,
                        const float* __restrict__ colsum, int* __restrict__ knn,
                        int N, int K) {
  __shared__ float sv[512];
  __shared__ int   si[512];
  const int n = blockIdx.x;
  for (int i = threadIdx.x; i < K; i += blockDim.x) {
    sv[i] = expf(h[(long)n * K + i] - colmax[i]) / colsum[i];
    si[i] = i;
  }
  __syncthreads();
  for (int k = 2; k <= K; k <<= 1) {
    for (int j = k >> 1; j > 0; j >>= 1) {
      for (int i = (int)threadIdx.x; i < K; i += blockDim.x) {
        const int ixj = i ^ j;
        if (ixj > i) {
          const float va = sv[i], vb = sv[ixj];
          const int ia = si[i], ib = si[ixj];
          const bool bFirst = pair_gt(vb, ib, va, ia);
          const bool doSwap = ((i & k) == 0) ? bFirst : !bFirst;
          if (doSwap) { sv[i] = vb; si[i] = ib; sv[ixj] = va; si[ixj] = ia; }
        }
      }
      __syncthreads();
    }
  }
  for (int i = threadIdx.x; i < 32; i += blockDim.x) knn[n * 32 + i] = si[i];
}

// X[(n*32+s), o] = leaky( fms[o, j] + fmq[o, n] + pos(dir) ), j = knn[n,s]
__global__ void k_crossX(const float* __restrict__ xyzq, const float* __restrict__ xyzs,
                         const float* __restrict__ fmq, const float* __restrict__ fms,
                         const int* __restrict__ knn, const float* __restrict__ pos_w,
                         const float* __restrict__ pos_b, float* __restrict__ X, int N) {
  const int g = blockIdx.x * blockDim.x + threadIdx.x;
  if (g >= N * 32 * 64) return;
  const int o = g & 63;
  const int s = (g >> 6) & 31;
  const int n = g >> 11;
  const int j = knn[n * 32 + s];
  const float d0 = xyzs[0 * N + j] - xyzq[0 * N + n];
  const float d1 = xyzs[1 * N + j] - xyzq[1 * N + n];
  const float d2 = xyzs[2 * N + j] - xyzq[2 * N + n];
  const float posd = pos_b[o] + pos_w[o * 3 + 0] * d0 + pos_w[o * 3 + 1] * d1 +
                     pos_w[o * 3 + 2] * d2;
  X[g] = leaky(fms[o * N + j] + fmq[o * N + n] + posd);
}

// out[o, n] = max_s Y[(n*32+s), o]
__global__ void k_maxred(const float* __restrict__ Y, float* __restrict__ out, int N) {
  const int g = blockIdx.x * blockDim.x + threadIdx.x;
  if (g >= N * 64) return;
  const int o = g & 63, n = g >> 6;
  float m = -3.402823466e+38f;
  for (int s = 0; s < 32; ++s) m = fmaxf(m, Y[((long)n * 32 + s) * 64 + o]);
  out[o * N + n] = m;
}

// ---------------------------------------------------------------------------

static void run_cross(const float* xyzq, const float* xyzs, const float* fmq,
                      const float* fms, const float* pos_w, const float* pos_b,
                      const float* mlp_w, const float* mlp_b, const float* const* wn,
                      float* corr, float* cvv, int* cii, float* hbuf, float* colmax,
                      float* colsum, int* knn, float* X, float* Y, float* outDN,
                      hipStream_t stream) {
  const int N = 2048, K = 512;
  k_corr<<<2048, 256, 0, stream>>>(fmq, fms, corr, N, 0.125f);   // 1/sqrt(64)
  k_top512<<<N, 256, 0, stream>>>(corr, cvv, cii, N, K);
  k_weightnet<<<(N * K) / 256, 256, 0, stream>>>(
      xyzq, xyzs, cvv, cii, wn[0], wn[1], wn[2], wn[3], wn[4], wn[5], wn[6], wn[7],
      hbuf, N, K);
  k_colstats<<<K, 256, 0, stream>>>(hbuf, colmax, colsum, N, K);
  k_top32<<<N, 256, 0, stream>>>(hbuf, colmax, colsum, knn, N, K);
  k_crossX<<<(N * 32 * 64) / 256, 256, 0, stream>>>(xyzq, xyzs, fmq, fms, knn, pos_w,
                                                    pos_b, X, N);
  k_mlp64<<<2048, 256, 0, stream>>>(X, mlp_w, mlp_b, Y, N * 32);  // 16384 tiles / 8
  k_maxred<<<(N * 64) / 256, 256, 0, stream>>>(Y, outDN, N);
}

extern "C" void kernel_launch(void* const* d_in, const int* in_sizes, int n_in,
                              void* d_out, int out_size, void* d_ws, size_t ws_size,
                              hipStream_t stream) {
  (void)in_sizes; (void)n_in; (void)out_size; (void)ws_size;
  const int N = 2048, K = 512;

  const float* pc1    = (const float*)d_in[0];
  const float* pc2    = (const float*)d_in[1];
  const float* feat1  = (const float*)d_in[2];
  const float* feat2  = (const float*)d_in[3];
  const float* w_t11  = (const float*)d_in[4];  const float* b_t11 = (const float*)d_in[5];
  const float* w_t22  = (const float*)d_in[6];  const float* b_t22 = (const float*)d_in[7];
  const float* w_pos1 = (const float*)d_in[8];  const float* b_pos1 = (const float*)d_in[9];
  const float* w_m1   = (const float*)d_in[10]; const float* b_m1 = (const float*)d_in[11];
  const float* w_t1   = (const float*)d_in[12]; const float* b_t1 = (const float*)d_in[13];
  const float* w_t2   = (const float*)d_in[14]; const float* b_t2 = (const float*)d_in[15];
  const float* w_pos2 = (const float*)d_in[16]; const float* b_pos2 = (const float*)d_in[17];
  const float* w_m2   = (const float*)d_in[18]; const float* b_m2 = (const float*)d_in[19];
  const float* wn[8]  = {(const float*)d_in[20], (const float*)d_in[21],
                         (const float*)d_in[22], (const float*)d_in[23],
                         (const float*)d_in[24], (const float*)d_in[25],
                         (const float*)d_in[26], (const float*)d_in[27]};

  char* ws = (char*)d_ws;
  size_t off = 0;
  auto alloc = [&](size_t bytes) -> void* {
    void* p = (void*)(ws + off);
    off += (bytes + 255) & ~(size_t)255;
    return p;
  };
  float* fmap1  = (float*)alloc((size_t)64 * N * 4);
  float* fmap2  = (float*)alloc((size_t)64 * N * 4);
  float* corr   = (float*)alloc((size_t)N * N * 4);
  float* cvv    = (float*)alloc((size_t)N * K * 4);
  int*   cii    = (int*)  alloc((size_t)N * K * 4);
  float* hbuf   = (float*)alloc((size_t)N * K * 4);
  float* colmax = (float*)alloc((size_t)K * 4);
  float* colsum = (float*)alloc((size_t)K * 4);
  int*   knn    = (int*)  alloc((size_t)N * 32 * 4);
  float* X      = (float*)alloc((size_t)N * 32 * 64 * 4);
  float* Y      = (float*)alloc((size_t)N * 32 * 64 * 4);
  float* cross  = (float*)alloc((size_t)64 * N * 4);

  float* out0 = (float*)d_out;         // feat1_new  [64, 2048]
  float* out1 = out0 + 64 * N;         // feat2_new
  float* out2 = out0 + 2 * 64 * N;     // feat1_final

  // ---- Pass 1: feat1_new -------------------------------------------------
  k_conv64<<<64, 256, 0, stream>>>(feat1, w_t11, b_t11, fmap1, N);
  k_conv64<<<64, 256, 0, stream>>>(feat2, w_t22, b_t22, fmap2, N);
  run_cross(pc1, pc2, fmap1, fmap2, w_pos1, b_pos1, w_m1, b_m1, wn, corr, cvv, cii,
            hbuf, colmax, colsum, knn, X, Y, cross, stream);
  k_conv64<<<64, 256, 0, stream>>>(cross, w_t1, b_t1, out0, N);

  // ---- Pass 2: feat2_new (reversed) --------------------------------------
  k_conv64<<<64, 256, 0, stream>>>(feat2, w_t11, b_t11, fmap1, N);
  k_conv64<<<64, 256, 0, stream>>>(feat1, w_t22, b_t22, fmap2, N);
  run_cross(pc2, pc1, fmap1, fmap2, w_pos1, b_pos1, w_m1, b_m1, wn, corr, cvv, cii,
            hbuf, colmax, colsum, knn, X, Y, cross, stream);
  k_conv64<<<64, 256, 0, stream>>>(cross, w_t2, b_t2, out1, N);

  // ---- Pass 3: feat1_final (no trailing conv) ----------------------------
  run_cross(pc1, pc2, out0, out1, w_pos2, b_pos2, w_m2, b_m2, wn, corr, cvv, cii,
            hbuf, colmax, colsum, knn, X, Y, out2, stream);
}